// UniPhyBlock_88579405513035
// MI455X (gfx1250) — compile-verified
//
#include <hip/hip_runtime.h>
#include <math.h>

#define DEVI __device__ __forceinline__

typedef float v2f __attribute__((ext_vector_type(2)));
typedef float v8f __attribute__((ext_vector_type(8)));
typedef unsigned int u32x4 __attribute__((ext_vector_type(4)));
typedef int i32x4 __attribute__((ext_vector_type(4)));
typedef int i32x8 __attribute__((ext_vector_type(8)));

constexpr int B_ = 2, T_ = 16, D_ = 64, H_ = 48, W_ = 48, E_ = 4, F_ = 128;
constexpr int C2 = 2 * D_;          // 128
constexpr int HW = H_ * W_;         // 2304
constexpr int BT = B_ * T_;         // 32
constexpr int ROWS = BT * HW;       // 73728

// ---------------- WMMA helper: D = A(16x4,f32) x B(4x16,f32) + C (16x16,f32) --
DEVI v8f wmma4(v2f a, v2f b, v8f c) {
  return __builtin_amdgcn_wmma_f32_16x16x4_f32(false, a, false, b, (short)0, c,
                                               false, false);
}

DEVI float sigmoidf_(float x) { return 1.f / (1.f + expf(-x)); }
DEVI float softplusf_(float x) { return (x > 20.f) ? x : log1pf(expf(x)); }
DEVI float gelu_tanh(float x) {
  const float k = 0.7978845608028654f; // sqrt(2/pi)
  return 0.5f * x * (1.f + tanhf(k * (x + 0.044715f * x * x * x)));
}

// -------- TDM: DMA a (rows x 16)-fp32 panel from global into LDS -------------
// Descriptor per CDNA5 ISA ch.8: group0 {count=1, lds_addr, global_addr, type=2},
// group1 {data_size=4B, tensor_dim0=16, tensor_dim1=rows, tile_dim0=16,
//         tile_dim1=rows, tensor_dim0_stride=stride}.
DEVI void stage_panel16(float* dst, const float* src, int rows, int stride) {
#if __has_builtin(__builtin_amdgcn_tensor_load_to_lds)
  // wave's previous LDS reads of this panel must retire before TDM overwrites
  __asm__ volatile("s_wait_dscnt 0x0" ::: "memory");
  unsigned long long ga = (unsigned long long)(const void*)src;
  u32x4 g0;
  g0.x = 1u;                                            // count = 1
  g0.y = (unsigned)(unsigned long long)(const void*)dst; // LDS byte offset
  g0.z = (unsigned)ga;
  g0.w = (unsigned)((ga >> 32) & 0x01ffffffull) | (2u << 30); // addr hi | type=2
  i32x8 g1;
  unsigned r = (unsigned)rows;
  g1[0] = (int)(2u << 16);                 // data_size = 4 bytes
  g1[1] = (int)(16u << 16);                // tensor_dim0 = 16
  g1[2] = (int)((r & 0xffffu) << 16);      // tensor_dim1 lo
  g1[3] = (int)(16u << 16);                // tile_dim0 = 16
  g1[4] = (int)(r & 0xffffu);              // tile_dim1 = rows, tile_dim2 = 0
  g1[5] = (int)stride;                     // tensor_dim0_stride (elements)
  g1[6] = 0;
  g1[7] = 0;
  i32x4 z4 = (i32x4)(0);
# if defined(__clang_major__) && (__clang_major__ >= 23)
  i32x8 z8 = (i32x8)(0);
  __builtin_amdgcn_tensor_load_to_lds(g0, g1, z4, z4, z8, 0);
# else
  __builtin_amdgcn_tensor_load_to_lds(g0, g1, z4, z4, 0);
# endif
#else
  int lane = (int)(threadIdx.x & 31);
  for (int i = lane; i < rows * 16; i += 32)
    dst[i] = src[(i >> 4) * stride + (i & 15)];
#endif
}

DEVI void stage_wait() {
#if __has_builtin(__builtin_amdgcn_s_wait_tensorcnt)
  __builtin_amdgcn_s_wait_tensorcnt(0);
#endif
  __asm__ volatile("" ::: "memory");
}

// ============ K1: spatial LN over 128 channels + metric scale -> xn ============
__global__ void k1_ln_spatial(const float* __restrict__ xre,
                              const float* __restrict__ xim,
                              const float* __restrict__ lw,
                              const float* __restrict__ lb,
                              const float* __restrict__ metric,
                              float* __restrict__ xn) {
  size_t g = (size_t)blockIdx.x * 256 + threadIdx.x; // over BT*HW
  int pix = (int)(g % HW);
  int bt  = (int)(g / HW);
  const float* pr = xre + (size_t)bt * D_ * HW + pix;
  const float* pi = xim + (size_t)bt * D_ * HW + pix;
  float s = 0.f, s2 = 0.f;
  for (int c = 0; c < 64; ++c) { float v = pr[(size_t)c * HW]; s += v; s2 += v * v; }
  for (int c = 0; c < 64; ++c) { float v = pi[(size_t)c * HW]; s += v; s2 += v * v; }
  float mean = s * (1.f / 128.f);
  float var  = s2 * (1.f / 128.f) - mean * mean;
  float rs   = rsqrtf(var + 1e-5f);
  float mt   = metric[pix];
  float* po = xn + (size_t)bt * C2 * HW + pix;
  for (int c = 0; c < C2; ++c) {
    float v = (c < 64) ? pr[(size_t)c * HW] : pi[(size_t)(c - 64) * HW];
    po[(size_t)c * HW] = ((v - mean) * rs * lw[c] + lb[c]) * mt;
  }
}

// ============ K2: 3x3 conv 128->128 via WMMA (implicit GEMM over 9 taps) ======
__global__ void k2_conv(const float* __restrict__ xn, const float* __restrict__ cw,
                        const float* __restrict__ cb, const float* __restrict__ xre,
                        const float* __restrict__ xim, float* __restrict__ xs_re,
                        float* __restrict__ xs_im) {
  int pt = blockIdx.x, ot = blockIdx.y, bt = blockIdx.z;
  int l = threadIdx.x & 31;
  int m = l & 15, half = l >> 4;
  int y0 = (pt * 16) / W_;           // W_=48 divisible by 16 -> no row wrap
  int x0 = (pt * 16) % W_;
  int ocA = ot * 16 + m;             // A-fragment row for this lane
  const float* xnf = xn + (size_t)bt * C2 * HW;
  v8f acc = (v8f)(0.0f);
  for (int ky = 0; ky < 3; ++ky) {
    int yy = y0 + ky - 1;
    bool yok = (yy >= 0) && (yy < H_);
    for (int kx = 0; kx < 3; ++kx) {
      int xb = x0 + m + kx - 1;      // B-fragment pixel column
      bool ok = yok && (xb >= 0) && (xb < W_);
      const float* ip0 = xnf + (size_t)(yy * W_ + xb);
      __builtin_prefetch(ip0, 0, 1);
      for (int kc = 0; kc < 32; ++kc) {
        int ka = kc * 4 + half * 2;
        const float* wp = cw + (((size_t)ocA * C2 + ka) * 3 + ky) * 3 + kx;
        v2f a; a.x = wp[0]; a.y = wp[9];
        v2f b;
        if (ok) {
          const float* ip = ip0 + (size_t)ka * HW;
          b.x = ip[0]; b.y = ip[HW];
        } else { b.x = 0.f; b.y = 0.f; }
        acc = wmma4(a, b, acc);
      }
    }
  }
#pragma unroll
  for (int v = 0; v < 8; ++v) {
    int oc = ot * 16 + v + half * 8;
    int pix = y0 * W_ + x0 + m;
    float val = acc[v] + cb[oc];
    size_t idx = ((size_t)bt * D_ + (oc & 63)) * HW + pix;
    if (oc < 64) xs_re[idx] = xre[idx] + val;
    else         xs_im[idx] = xim[idx] + val;
  }
}

// ============ K3: complex encode  x_eig = xs @ E_enc ==========================
__global__ void k3_encode(const float* __restrict__ xs_re,
                          const float* __restrict__ xs_im,
                          const float* __restrict__ Er,
                          const float* __restrict__ Ei,
                          float* __restrict__ xe_re, float* __restrict__ xe_im) {
  int rt = blockIdx.x;
  int l = threadIdx.x & 31;
  int m = l & 15, half = l >> 4;
  int row = rt * 16 + m;
  int bt = row / HW, pix = row % HW;
  const float* pre = xs_re + (size_t)bt * D_ * HW + pix;
  const float* pim = xs_im + (size_t)bt * D_ * HW + pix;
  // register-cache A fragments (reused by all 4 column tiles)
  v2f Arr[16], Aii[16];
#pragma unroll
  for (int kc = 0; kc < 16; ++kc) {
    int k0 = kc * 4 + half * 2;
    Arr[kc].x = pre[(size_t)k0 * HW]; Arr[kc].y = pre[(size_t)(k0 + 1) * HW];
    Aii[kc].x = pim[(size_t)k0 * HW]; Aii[kc].y = pim[(size_t)(k0 + 1) * HW];
  }
  for (int ct = 0; ct < 4; ++ct) {
    int e = ct * 16 + m;
    v8f ar = (v8f)(0.0f), ai = (v8f)(0.0f);
#pragma unroll
    for (int kc = 0; kc < 16; ++kc) {
      int k0 = kc * 4 + half * 2;
      v2f Br, Bi, nAi;
      nAi.x = -Aii[kc].x; nAi.y = -Aii[kc].y;
      Br.x = Er[k0 * 64 + e]; Br.y = Er[(k0 + 1) * 64 + e];
      Bi.x = Ei[k0 * 64 + e]; Bi.y = Ei[(k0 + 1) * 64 + e];
      ar = wmma4(Arr[kc], Br, ar); ar = wmma4(nAi, Bi, ar);
      ai = wmma4(Arr[kc], Bi, ai); ai = wmma4(Aii[kc], Br, ai);
    }
#pragma unroll
    for (int v = 0; v < 8; ++v) {
      int ro = rt * 16 + v + half * 8;
      xe_re[(size_t)ro * 64 + ct * 16 + m] = ar[v];
      xe_im[(size_t)ro * 64 + ct * 16 + m] = ai[v];
    }
  }
}

// ============ K4: x_mean over pixels -> (BT, D) ===============================
__global__ void k4_xmean(const float* __restrict__ xe_re,
                         const float* __restrict__ xe_im,
                         float* __restrict__ xm_re, float* __restrict__ xm_im) {
  int bt = blockIdx.x >> 6, e = blockIdx.x & 63;
  __shared__ float sr[256], si[256];
  float ar = 0.f, ai = 0.f;
  for (int p = threadIdx.x; p < HW; p += 256) {
    size_t idx = ((size_t)bt * HW + p) * 64 + e;
    ar += xe_re[idx]; ai += xe_im[idx];
  }
  sr[threadIdx.x] = ar; si[threadIdx.x] = ai;
  __syncthreads();
  for (int off = 128; off > 0; off >>= 1) {
    if ((int)threadIdx.x < off) {
      sr[threadIdx.x] += sr[threadIdx.x + off];
      si[threadIdx.x] += si[threadIdx.x + off];
    }
    __syncthreads();
  }
  if (threadIdx.x == 0) {
    xm_re[bt * 64 + e] = sr[0] * (1.f / HW);
    xm_im[bt * 64 + e] = si[0] * (1.f / HW);
  }
}

// ============ K5: lam, op_decay, op_forcing per (t,d) =========================
__global__ void k5_ops(const float* __restrict__ dt, const float* __restrict__ lam_nu,
                       const float* __restrict__ lam_im, float* __restrict__ odr,
                       float* __restrict__ odi, float* __restrict__ ofr,
                       float* __restrict__ ofi) {
  int t = threadIdx.x >> 6, d = threadIdx.x & 63;  // 1024 threads
  float lr = -softplusf_(lam_nu[d]) - 1e-4f;
  float li = lam_im[d];
  float dts = dt[t];
  float er = expf(lr * dts);
  float dr = er * cosf(li * dts), di = er * sinf(li * dts);
  float nr = dr - 1.f, ni = di;
  float den = lr * lr + li * li;
  float fr = (nr * lr + ni * li) / den;
  float fi = (ni * lr - nr * li) / den;
  odr[t * 64 + d] = dr; odi[t * 64 + d] = di;
  ofr[t * 64 + d] = fr; ofi[t * 64 + d] = fi;
}

// ============ K6: sequential flux scan over T (B*D = 128 threads) =============
__global__ void k6_flux(const float* __restrict__ xm_re, const float* __restrict__ xm_im,
                        const float* __restrict__ fp_re, const float* __restrict__ fp_im,
                        const float* __restrict__ Wg, const float* __restrict__ bg,
                        const float* __restrict__ al_re, const float* __restrict__ al_im,
                        float* __restrict__ gseq, float* __restrict__ src_re,
                        float* __restrict__ src_im) {
  __shared__ float smr[2][64];
  int b = threadIdx.x >> 6, d = threadIdx.x & 63;
  float fr = fp_re[b * 64 + d], fi = fp_im[b * 64 + d];
  float ar = al_re[d], ai = al_im[d];
  for (int t = 0; t < T_; ++t) {
    int bt = b * T_ + t;
    smr[b][d] = xm_re[(size_t)bt * 64 + d];
    __syncthreads();
    float acc = bg[d];
    for (int k = 0; k < 64; ++k) acc += smr[b][k] * Wg[k * 64 + d];
    float g = sigmoidf_(acc);
    float xr = xm_re[(size_t)bt * 64 + d];
    float xi = xm_im[(size_t)bt * 64 + d];
    fr = fr * g + xr * (1.f - g);
    fi = fi * g + xi * (1.f - g);
    gseq[bt * 64 + d] = g;
    src_re[bt * 64 + d] = fr * ar - fi * ai;
    src_im[bt * 64 + d] = fr * ai + fi * ar;
    __syncthreads();
  }
}

// ============ K7: forcing + temporal scan (in-place x_eig -> u) ===============
__global__ void k7_scan(const float* __restrict__ gseq, const float* __restrict__ src_re,
                        const float* __restrict__ src_im, const float* __restrict__ odr,
                        const float* __restrict__ odi, const float* __restrict__ ofr,
                        const float* __restrict__ ofi, const float* __restrict__ hp_re,
                        const float* __restrict__ hp_im, float* __restrict__ xe_re,
                        float* __restrict__ xe_im) {
  size_t g = (size_t)blockIdx.x * 256 + threadIdx.x; // B*HW*64
  int d = (int)(g & 63);
  size_t pix = (g >> 6) % HW;
  int b = (int)(g / ((size_t)64 * HW));
  float hpr = hp_re[((size_t)b * HW + pix) * 64 + d];
  float hpi = hp_im[((size_t)b * HW + pix) * 64 + d];
  float hr = 0.f, hi = 0.f;
  for (int t = 0; t < T_; ++t) {
    int bt = b * T_ + t;
    size_t idx = ((size_t)bt * HW + pix) * 64 + d;
    float xr = xe_re[idx], xi = xe_im[idx];
    float gg = gseq[bt * 64 + d];
    float sr = src_re[bt * 64 + d], si = src_im[bt * 64 + d];
    float fr0 = xr * gg + sr * (1.f - gg);
    float fi0 = xi * gg + si * (1.f - gg);
    float pfr = ofr[t * 64 + d], pfi = ofi[t * 64 + d];
    float ur = fr0 * pfr - fi0 * pfi;
    float ui = fr0 * pfi + fi0 * pfr;
    float Ar = odr[t * 64 + d], Ai = odi[t * 64 + d];
    float nhr = Ar * hr - Ai * hi + ur;
    float nhi = Ar * hi + Ai * hr + ui;
    hr = nhr; hi = nhi;
    xe_re[idx] = hr + hpr * Ar - hpi * Ai;
    xe_im[idx] = hi + hpr * Ai + hpi * Ar;
  }
}

// ============ K8: complex decode  xo = u @ E_dec ==============================
__global__ void k8_decode(const float* __restrict__ u_re, const float* __restrict__ u_im,
                          const float* __restrict__ Er, const float* __restrict__ Ei,
                          float* __restrict__ xo_re, float* __restrict__ xo_im) {
  int rt = blockIdx.x;
  int l = threadIdx.x & 31;
  int m = l & 15, half = l >> 4;
  const float* pr = u_re + ((size_t)(rt * 16 + m)) * 64;
  const float* pi = u_im + ((size_t)(rt * 16 + m)) * 64;
  v2f Arr[16], Aii[16];
#pragma unroll
  for (int kc = 0; kc < 16; ++kc) {
    int k0 = kc * 4 + half * 2;
    Arr[kc].x = pr[k0]; Arr[kc].y = pr[k0 + 1];
    Aii[kc].x = pi[k0]; Aii[kc].y = pi[k0 + 1];
  }
  for (int ct = 0; ct < 4; ++ct) {
    int e = ct * 16 + m;
    v8f ar = (v8f)(0.0f), ai = (v8f)(0.0f);
#pragma unroll
    for (int kc = 0; kc < 16; ++kc) {
      int k0 = kc * 4 + half * 2;
      v2f Br, Bi, nAi;
      nAi.x = -Aii[kc].x; nAi.y = -Aii[kc].y;
      Br.x = Er[k0 * 64 + e]; Br.y = Er[(k0 + 1) * 64 + e];
      Bi.x = Ei[k0 * 64 + e]; Bi.y = Ei[(k0 + 1) * 64 + e];
      ar = wmma4(Arr[kc], Br, ar); ar = wmma4(nAi, Bi, ar);
      ai = wmma4(Arr[kc], Bi, ai); ai = wmma4(Aii[kc], Br, ai);
    }
#pragma unroll
    for (int v = 0; v < 8; ++v) {
      int ro = rt * 16 + v + half * 8;
      xo_re[(size_t)ro * 64 + ct * 16 + m] = ar[v];
      xo_im[(size_t)ro * 64 + ct * 16 + m] = ai[v];
    }
  }
}

// ============ K9: temporal LN + softmax gates (one wave per row) ==============
__global__ void k9_lnt(const float* __restrict__ xo_re, const float* __restrict__ xo_im,
                       const float* __restrict__ lw, const float* __restrict__ lb,
                       const float* __restrict__ Wgate, float* __restrict__ xc_re,
                       float* __restrict__ xc_im, float* __restrict__ gates) {
  int wid = threadIdx.x >> 5;
  int l = threadIdx.x & 31;
  size_t row = (size_t)blockIdx.x * 8 + wid;
  const float* pr = xo_re + row * 64;
  const float* pi = xo_im + row * 64;
  int c0 = l * 4;
  float v0[4];
#pragma unroll
  for (int j = 0; j < 4; ++j) {
    int c = c0 + j;
    v0[j] = (c < 64) ? pr[c] : pi[c - 64];
  }
  float s = 0.f, s2 = 0.f;
#pragma unroll
  for (int j = 0; j < 4; ++j) { s += v0[j]; s2 += v0[j] * v0[j]; }
  for (int off = 16; off > 0; off >>= 1) {
    s  += __shfl_xor(s, off, 32);
    s2 += __shfl_xor(s2, off, 32);
  }
  float mean = s * (1.f / 128.f);
  float var  = s2 * (1.f / 128.f) - mean * mean;
  float rs   = rsqrtf(var + 1e-5f);
  float nv[4];
#pragma unroll
  for (int j = 0; j < 4; ++j) {
    int c = c0 + j;
    nv[j] = (v0[j] - mean) * rs * lw[c] + lb[c];
    if (c < 64) xc_re[row * 64 + c] = nv[j];
    else        xc_im[row * 64 + (c - 64)] = nv[j];
  }
  float p[4] = {0.f, 0.f, 0.f, 0.f};
  if (l < 16) {
#pragma unroll
    for (int j = 0; j < 4; ++j)
#pragma unroll
      for (int e = 0; e < 4; ++e) p[e] += nv[j] * Wgate[(c0 + j) * E_ + e];
  }
  for (int off = 16; off > 0; off >>= 1)
#pragma unroll
    for (int e = 0; e < 4; ++e) p[e] += __shfl_xor(p[e], off, 32);
  float mx = fmaxf(fmaxf(p[0], p[1]), fmaxf(p[2], p[3]));
  float ex[4], se = 0.f;
#pragma unroll
  for (int e = 0; e < 4; ++e) { ex[e] = expf(p[e] - mx); se += ex[e]; }
  if (l == 0) {
#pragma unroll
    for (int e = 0; e < 4; ++e) gates[row * E_ + e] = ex[e] / se;
  }
}

// ============ K10: per-expert complex MLP (WMMA + TDM weight staging) =========
__global__ void k10_mlp(const float* __restrict__ xc_re, const float* __restrict__ xc_im,
                        const float* __restrict__ gates, const float* __restrict__ W1r,
                        const float* __restrict__ W1i, const float* __restrict__ W2r,
                        const float* __restrict__ W2i, float* __restrict__ xf_re,
                        float* __restrict__ xf_im) {
  __shared__ float aR[16 * F_];
  __shared__ float aI[16 * F_];
  __shared__ float pR[128 * 16];   // staged weight panel (re), max 128x16
  __shared__ float pI[128 * 16];   // staged weight panel (im)
  __shared__ float gl[16 * E_];
  int rt = blockIdx.x;
  int l = threadIdx.x & 31;
  int m = l & 15, half = l >> 4;
  if (l < 16) {
#pragma unroll
    for (int e = 0; e < E_; ++e) gl[l * E_ + e] = gates[(size_t)(rt * 16 + l) * E_ + e];
  }
  __syncthreads();
  const float* pxr = xc_re + (size_t)(rt * 16 + m) * 64;
  const float* pxi = xc_im + (size_t)(rt * 16 + m) * 64;
  // register-cache A fragments of the xc row-tile (reused by all e, ct)
  v2f Xr[16], Xi[16];
#pragma unroll
  for (int kc = 0; kc < 16; ++kc) {
    int k0 = kc * 4 + half * 2;
    Xr[kc].x = pxr[k0]; Xr[kc].y = pxr[k0 + 1];
    Xi[kc].x = pxi[k0]; Xi[kc].y = pxi[k0 + 1];
  }
  v8f dR[4], dI[4];
#pragma unroll
  for (int c = 0; c < 4; ++c) { dR[c] = (v8f)(0.0f); dI[c] = (v8f)(0.0f); }

  for (int e = 0; e < E_; ++e) {
    const float* w1r = W1r + (size_t)e * D_ * F_;
    const float* w1i = W1i + (size_t)e * D_ * F_;
    const float* w2r = W2r + (size_t)e * F_ * D_;
    const float* w2i = W2i + (size_t)e * F_ * D_;
    // ---- GEMM1: a = gelu(xc @ W1[e])  [16x64]x[64x128] ----
    for (int ct = 0; ct < 8; ++ct) {
      stage_panel16(pR, w1r + ct * 16, 64, F_);   // 64x16 panel via TDM
      stage_panel16(pI, w1i + ct * 16, 64, F_);
      stage_wait();
      int col = ct * 16 + m;
      v8f hr = (v8f)(0.0f), hi = (v8f)(0.0f);
#pragma unroll
      for (int kc = 0; kc < 16; ++kc) {
        int k0 = kc * 4 + half * 2;
        v2f Br, Bi, nAi;
        nAi.x = -Xi[kc].x; nAi.y = -Xi[kc].y;
        Br.x = pR[k0 * 16 + m]; Br.y = pR[(k0 + 1) * 16 + m];
        Bi.x = pI[k0 * 16 + m]; Bi.y = pI[(k0 + 1) * 16 + m];
        hr = wmma4(Xr[kc], Br, hr); hr = wmma4(nAi, Bi, hr);
        hi = wmma4(Xr[kc], Bi, hi); hi = wmma4(Xi[kc], Br, hi);
      }
#pragma unroll
      for (int v = 0; v < 8; ++v) {
        int rm = v + half * 8;
        aR[rm * F_ + col] = gelu_tanh(hr[v]);
        aI[rm * F_ + col] = gelu_tanh(hi[v]);
      }
    }
    __syncthreads();
    // ---- GEMM2: h2 = a @ W2[e]  [16x128]x[128x64], gate-scaled accumulate ----
    for (int ct = 0; ct < 4; ++ct) {
      stage_panel16(pR, w2r + ct * 16, 128, D_);  // 128x16 panel via TDM
      stage_panel16(pI, w2i + ct * 16, 128, D_);
      stage_wait();
      int col = ct * 16 + m;
      v8f hr = (v8f)(0.0f), hi = (v8f)(0.0f);
#pragma unroll
      for (int kc = 0; kc < 32; ++kc) {
        int k0 = kc * 4 + half * 2;
        v2f Ar, Ai, nAi, Br, Bi;
        Ar.x = aR[m * F_ + k0]; Ar.y = aR[m * F_ + k0 + 1];
        Ai.x = aI[m * F_ + k0]; Ai.y = aI[m * F_ + k0 + 1];
        nAi.x = -Ai.x; nAi.y = -Ai.y;
        Br.x = pR[k0 * 16 + m]; Br.y = pR[(k0 + 1) * 16 + m];
        Bi.x = pI[k0 * 16 + m]; Bi.y = pI[(k0 + 1) * 16 + m];
        hr = wmma4(Ar, Br, hr); hr = wmma4(nAi, Bi, hr);
        hi = wmma4(Ar, Bi, hi); hi = wmma4(Ai, Br, hi);
      }
#pragma unroll
      for (int v = 0; v < 8; ++v) {
        int rm = v + half * 8;
        float g = gl[rm * E_ + e];
        dR[ct][v] += g * hr[v];
        dI[ct][v] += g * hi[v];
      }
    }
    __syncthreads();
  }
#pragma unroll
  for (int ct = 0; ct < 4; ++ct) {
#pragma unroll
    for (int v = 0; v < 8; ++v) {
      int rm = v + half * 8;
      size_t idx = (size_t)(rt * 16 + rm) * 64 + ct * 16 + m;
      xf_re[idx] = xc_re[idx] + dR[ct][v];
      xf_im[idx] = xc_im[idx] + dI[ct][v];
    }
  }
}

// ============ K11: z = xs + xo2 ; stack (re, im) last axis ====================
__global__ void k11_out(const float* __restrict__ xs_re, const float* __restrict__ xs_im,
                        const float* __restrict__ xf_re, const float* __restrict__ xf_im,
                        float* __restrict__ out) {
  size_t g = (size_t)blockIdx.x * 256 + threadIdx.x; // BT*D*HW
  size_t pix = g % HW;
  int d = (int)((g / HW) & 63);
  int bt = (int)(g / ((size_t)HW * 64));
  size_t xidx = ((size_t)bt * 64 + d) * HW + pix;
  size_t ridx = ((size_t)bt * HW + pix) * 64 + d;
  out[xidx * 2]     = xs_re[xidx] + xf_re[ridx];
  out[xidx * 2 + 1] = xs_im[xidx] + xf_im[ridx];
}

// =============================================================================
extern "C" void kernel_launch(void* const* d_in, const int* in_sizes, int n_in,
                              void* d_out, int out_size, void* d_ws, size_t ws_size,
                              hipStream_t stream) {
  (void)in_sizes; (void)n_in; (void)out_size; (void)ws_size;
  const float* x_re      = (const float*)d_in[0];
  const float* x_im      = (const float*)d_in[1];
  const float* h_prev_re = (const float*)d_in[2];
  const float* h_prev_im = (const float*)d_in[3];
  const float* dt        = (const float*)d_in[4];
  const float* flux_re   = (const float*)d_in[5];
  const float* flux_im   = (const float*)d_in[6];
  const float* ln_sp_w   = (const float*)d_in[7];
  const float* ln_sp_b   = (const float*)d_in[8];
  const float* conv_w    = (const float*)d_in[9];
  const float* conv_b    = (const float*)d_in[10];
  const float* metric    = (const float*)d_in[11];
  const float* ln_t_w    = (const float*)d_in[12];
  const float* ln_t_b    = (const float*)d_in[13];
  const float* E_enc_re  = (const float*)d_in[14];
  const float* E_enc_im  = (const float*)d_in[15];
  const float* E_dec_re  = (const float*)d_in[16];
  const float* E_dec_im  = (const float*)d_in[17];
  const float* Wg        = (const float*)d_in[18];
  const float* bg        = (const float*)d_in[19];
  const float* alpha_re  = (const float*)d_in[20];
  const float* alpha_im  = (const float*)d_in[21];
  const float* lam_nu    = (const float*)d_in[22];
  const float* lam_im    = (const float*)d_in[23];
  const float* Wgate     = (const float*)d_in[24];
  const float* W1_re     = (const float*)d_in[25];
  const float* W1_im     = (const float*)d_in[26];
  const float* W2_re     = (const float*)d_in[27];
  const float* W2_im     = (const float*)d_in[28];

  float* w = (float*)d_ws;
  size_t o = 0;
  float* XN  = w + o; o += (size_t)BT * C2 * HW;   // also reused as xo
  float* XSR = w + o; o += (size_t)BT * D_ * HW;
  float* XSI = w + o; o += (size_t)BT * D_ * HW;
  float* XER = w + o; o += (size_t)ROWS * D_;      // x_eig -> u (in place) -> xcf
  float* XEI = w + o; o += (size_t)ROWS * D_;
  float* XCR = w + o; o += (size_t)ROWS * D_;
  float* XCI = w + o; o += (size_t)ROWS * D_;
  float* GTS = w + o; o += (size_t)ROWS * E_;
  float* XMR = w + o; o += BT * D_;
  float* XMI = w + o; o += BT * D_;
  float* GSQ = w + o; o += BT * D_;
  float* SRR = w + o; o += BT * D_;
  float* SRI = w + o; o += BT * D_;
  float* ODR = w + o; o += T_ * D_;
  float* ODI = w + o; o += T_ * D_;
  float* OFR = w + o; o += T_ * D_;
  float* OFI = w + o; o += T_ * D_;
  float* XOR_ = XN;
  float* XOI_ = XN + (size_t)ROWS * D_;
  float* XFR = XER;
  float* XFI = XEI;

  k1_ln_spatial<<<(BT * HW) / 256, 256, 0, stream>>>(x_re, x_im, ln_sp_w, ln_sp_b,
                                                     metric, XN);
  k2_conv<<<dim3(HW / 16, C2 / 16, BT), 32, 0, stream>>>(XN, conv_w, conv_b, x_re,
                                                         x_im, XSR, XSI);
  k3_encode<<<ROWS / 16, 32, 0, stream>>>(XSR, XSI, E_enc_re, E_enc_im, XER, XEI);
  k4_xmean<<<BT * D_, 256, 0, stream>>>(XER, XEI, XMR, XMI);
  k5_ops<<<1, T_ * D_, 0, stream>>>(dt, lam_nu, lam_im, ODR, ODI, OFR, OFI);
  k6_flux<<<1, B_ * D_, 0, stream>>>(XMR, XMI, flux_re, flux_im, Wg, bg, alpha_re,
                                     alpha_im, GSQ, SRR, SRI);
  k7_scan<<<(B_ * HW * D_) / 256, 256, 0, stream>>>(GSQ, SRR, SRI, ODR, ODI, OFR,
                                                    OFI, h_prev_re, h_prev_im, XER,
                                                    XEI);
  k8_decode<<<ROWS / 16, 32, 0, stream>>>(XER, XEI, E_dec_re, E_dec_im, XOR_, XOI_);
  k9_lnt<<<ROWS / 8, 256, 0, stream>>>(XOR_, XOI_, ln_t_w, ln_t_b, Wgate, XCR, XCI,
                                       GTS);
  k10_mlp<<<ROWS / 16, 32, 0, stream>>>(XCR, XCI, GTS, W1_re, W1_im, W2_re, W2_im,
                                        XFR, XFI);
  k11_out<<<(BT * D_ * HW) / 256, 256, 0, stream>>>(XSR, XSI, XFR, XFI,
                                                    (float*)d_out);
}